// ZetaLSTMConstrained_71433896067264
// MI455X (gfx1250) — compile-verified
//
#include <hip/hip_runtime.h>
#include <math.h>

// ---------------------------------------------------------------------------
// Zeta-LSTM for MI455X (gfx1250, wave32).
//   outputs [B,T,H] ++ h_f [1,B,H] ++ c_f [1,B,H]  (fp32)
// Strategy:
//   k_wts   : w_t = sum_m softplus(log_phi)_m * cos(t*gamma_m) / 15
//   k_xproj : x_proj = x @ W_ih^T + b_ih + b_hh   (WMMA f32 16x16x4, parallel)
//   k_scan  : persistent 8-block kernel, 1 grid barrier / timestep,
//             gates = x_proj_t + h @ W_hh^T, fused LSTM cell + zeta term.
//             LDS double-buffered K-chunk pipeline + next-step prefetch.
// ---------------------------------------------------------------------------

typedef __attribute__((ext_vector_type(2))) float v2f;
typedef __attribute__((ext_vector_type(8))) float v8f;

static constexpr int Bc = 64;      // batch
static constexpr int Tc = 1024;    // time
static constexpr int Ic = 512;     // input
static constexpr int Hc = 512;     // hidden
static constexpr int G4 = 4 * Hc;  // 2048 gate columns
static constexpr int NBLK = 8;     // persistent scan blocks (Hc / 64)
static constexpr int LDSW = 132;   // padded LDS row stride (16B aligned, bank-spread)
static constexpr float ZW = 0.4f;

// ---------------- small helpers --------------------------------------------

__device__ __forceinline__ float sigm(float x) { return 1.0f / (1.0f + __expf(-x)); }

// Device-wide barrier: monotonically increasing counter, one signal per block.
__device__ __forceinline__ void grid_barrier(unsigned* cnt, unsigned target) {
    __syncthreads();
    __threadfence();                       // release our h/c writes (agent scope)
    if (threadIdx.x == 0) {
        atomicAdd(cnt, 1u);
        while (__hip_atomic_load(cnt, __ATOMIC_ACQUIRE, __HIP_MEMORY_SCOPE_AGENT) < target) {
            __builtin_amdgcn_s_sleep(2);
        }
    }
    __syncthreads();
    __threadfence();                       // acquire other blocks' writes
}

// Gather one [64 x 128] f32 tile into registers (512 thr, 4x float4 each).
__device__ __forceinline__ void ld_chunk(float4 stg[4], const float* __restrict__ src,
                                         size_t rowStride, int k0) {
#pragma unroll
    for (int r = 0; r < 4; ++r) {
        int j   = (int)threadIdx.x + 512 * r;   // 0..2047 float4 slots
        int row = j >> 5;                       // 32 float4 per row
        int c4  = j & 31;
        stg[r] = *(const float4*)(src + (size_t)row * rowStride + k0 + c4 * 4);
    }
}

// Spill staged registers into an LDS tile (padded stride).
__device__ __forceinline__ void st_chunk(float* lds, const float4 stg[4]) {
#pragma unroll
    for (int r = 0; r < 4; ++r) {
        int j   = (int)threadIdx.x + 512 * r;
        int row = j >> 5;
        int c4  = j & 31;
        *(float4*)&lds[row * LDSW + c4 * 4] = stg[r];
    }
}

// 32 K-steps (K=128) of the 4 gate tiles for one wave.
//  A frag: lane l -> M=l&15, K = 2*(l>>4) + vgpr   -> contiguous float2 in LDS.
//  B frag: lane l -> N=l&15, K = 2*(l>>4) + vgpr   -> contiguous float2 of W row.
__device__ __forceinline__ void mma_4x32(const float* __restrict__ hsA,
                                         const float* const Wb[4],
                                         int koff, int lh, v8f acc[4]) {
#pragma unroll 4
    for (int kk = 0; kk < 32; ++kk) {
        int ka = kk * 4 + lh * 2;
        v2f a = *(const v2f*)(hsA + ka);
#pragma unroll
        for (int g = 0; g < 4; ++g) {
            v2f b = *(const v2f*)(Wb[g] + koff + ka);
            acc[g] = __builtin_amdgcn_wmma_f32_16x16x4_f32(
                false, a, false, b, (short)0, acc[g], false, false);
        }
    }
}

// One K=512 GEMM phase with double-buffered LDS staging: global loads for
// chunk s+1 are issued before the WMMAs of chunk s (HW overlaps them); only
// ds_store + one __syncthreads per chunk sits on the critical path. Writers
// only ever touch the buffer no wave is reading -> one barrier is sufficient.
__device__ __forceinline__ void gemm_phase(float (*hs2)[64 * LDSW],
                                           const float* __restrict__ src, size_t rowStride,
                                           const float* const Wb[4],
                                           int arowOff, int lh, v8f acc[4]) {
    float4 stg[4];
    ld_chunk(stg, src, rowStride, 0);
    st_chunk(hs2[0], stg);
    __syncthreads();
#pragma unroll
    for (int s = 0; s < 4; ++s) {
        if (s < 3) ld_chunk(stg, src, rowStride, (s + 1) * 128);   // overlap w/ WMMA
        mma_4x32(hs2[s & 1] + arowOff, Wb, s * 128, lh, acc);
        if (s < 3) {
            st_chunk(hs2[(s + 1) & 1], stg);
            __syncthreads();
        }
    }
}

// ---------------- kernel 1: zeta mixing weights ----------------------------

__global__ void k_wts(const float* __restrict__ log_phi,
                      const float* __restrict__ gammas,
                      float* __restrict__ wts) {
    int t = blockIdx.x * blockDim.x + threadIdx.x;
    if (t >= Tc) return;
    float acc = 0.0f;
#pragma unroll
    for (int m = 0; m < 15; ++m) {
        float lp  = log_phi[m];
        float phi = (lp > 20.0f) ? lp : log1pf(expf(lp));   // softplus
        acc += phi * cosf((float)t * gammas[m]);
    }
    wts[t] = acc / 15.0f;
}

// ---------------- kernel 2: input projection GEMM --------------------------
// x_proj[bt, gc] = sum_k x[bt,k]*W_ih[gc,k] + b_ih[gc] + b_hh[gc]
// Block: 256 thr = 8 waves, tile 64(M) x 128(N). Grid: (65536/64, 2048/128).
// x (134 MB) and W_ih (4 MB) live in the 192 MB L2; compiler pipelines the
// v2f fragment loads against the WMMAs (seen in round-1 asm).

__global__ __launch_bounds__(256) void k_xproj(const float* __restrict__ x,
                                               const float* __restrict__ Wih,
                                               const float* __restrict__ bih,
                                               const float* __restrict__ bhh,
                                               float* __restrict__ xp) {
    const int wave = threadIdx.x >> 5, lane = threadIdx.x & 31;
    const int l15 = lane & 15, lh = lane >> 4;
    const int mt = wave & 3;        // M sub-tile
    const int nh = wave >> 2;       // N half
    const int row0 = blockIdx.x * 64 + mt * 16;
    const float* xA = x + (size_t)(row0 + l15) * Ic;

    v8f acc[4];
    int cols[4];
#pragma unroll
    for (int q = 0; q < 4; ++q) {
        int col = blockIdx.y * 128 + nh * 64 + q * 16 + l15;
        cols[q] = col;
        float bias = bih[col] + bhh[col];
#pragma unroll
        for (int v = 0; v < 8; ++v) acc[q][v] = bias;
    }
#pragma unroll 4
    for (int kk = 0; kk < Ic / 4; ++kk) {
        int ka = kk * 4 + lh * 2;
        v2f a = *(const v2f*)(xA + ka);
#pragma unroll
        for (int q = 0; q < 4; ++q) {
            v2f b = *(const v2f*)(Wih + (size_t)cols[q] * Ic + ka);
            acc[q] = __builtin_amdgcn_wmma_f32_16x16x4_f32(
                false, a, false, b, (short)0, acc[q], false, false);
        }
    }
#pragma unroll
    for (int q = 0; q < 4; ++q)
#pragma unroll
        for (int v = 0; v < 8; ++v) {
            int row = row0 + v + 8 * lh;                 // C/D: M = v + 8*(lane>=16)
            xp[(size_t)row * G4 + cols[q]] = acc[q][v];
        }
}

// ---------------- kernel 3: persistent recurrent scan ----------------------
// 8 blocks x 512 thr (16 waves). Block bx owns h-cols [64bx,64bx+64).
// Wave w: mt = w&3 (batch 16-rows), cg = w>>2 (col 16-group); its 4
// accumulators are the i/f/g/o gate tiles of the SAME 16x16 region, so the
// cell update is wave-local.

__global__ __launch_bounds__(512) void k_scan(const float* __restrict__ xp,  // may be null
                                              const float* __restrict__ x,
                                              const float* __restrict__ Wih,
                                              const float* __restrict__ Whh,
                                              const float* __restrict__ bih,
                                              const float* __restrict__ bhh,
                                              const float* __restrict__ wts,
                                              float* __restrict__ hA,
                                              float* __restrict__ hB,
                                              float* __restrict__ cbuf,
                                              float* __restrict__ out,
                                              unsigned* __restrict__ bar) {
    __shared__ float hs[2][64 * LDSW];   // 2 x 33 KB ping-pong (of 320 KB WGP LDS)

    const int bx = blockIdx.x;
    const int wave = threadIdx.x >> 5, lane = threadIdx.x & 31;
    const int l15 = lane & 15, lh = lane >> 4;
    const int mt = wave & 3, cg = wave >> 2;
    const int colL = bx * 64 + cg * 16 + l15;         // this lane's h column
    const int arowOff = (mt * 16 + l15) * LDSW;       // this lane's A row in LDS

    // Per-lane W row pointers for the 4 gates (L2-resident, 8 MB total).
    const float* Whb[4];
    const float* Wib[4];
    float bias[4];
#pragma unroll
    for (int g = 0; g < 4; ++g) {
        int gc = g * Hc + colL;
        Whb[g] = Whh + (size_t)gc * Hc;
        Wib[g] = Wih + (size_t)gc * Ic;
        bias[g] = bih[gc] + bhh[gc];
    }

    float* hb[2] = {hA, hB};
    unsigned tgt = 0;

    for (int t = 0; t < Tc; ++t) {
        const float* hcur = hb[t & 1];
        float* hnxt = hb[(t + 1) & 1];

        v8f acc[4];
        if (xp != nullptr) {
            // accumulators start from precomputed x-projection (+biases)
#pragma unroll
            for (int g = 0; g < 4; ++g)
#pragma unroll
                for (int v = 0; v < 8; ++v) {
                    int b = mt * 16 + v + 8 * lh;
                    acc[g][v] = xp[((size_t)b * Tc + t) * G4 + g * Hc + colL];
                }
        } else {
            // fused fallback: start from biases, add x_t @ W_ih^T here
#pragma unroll
            for (int g = 0; g < 4; ++g)
#pragma unroll
                for (int v = 0; v < 8; ++v) acc[g][v] = bias[g];
            gemm_phase(hs, x + (size_t)t * Ic, (size_t)Tc * Ic, Wib, arowOff, lh, acc);
        }

        // recurrent part: gates += h @ W_hh^T (K=512, double-buffered pipeline)
        gemm_phase(hs, hcur, (size_t)Hc, Whb, arowOff, lh, acc);

        // LSTM cell + zeta memory (all operands wave-local)
        float wt = wts[t];
#pragma unroll
        for (int v = 0; v < 8; ++v) {
            int b = mt * 16 + v + 8 * lh;
            size_t idx = (size_t)b * Hc + colL;
            float ig = acc[0][v], fg = acc[1][v], gg = acc[2][v], og = acc[3][v];
            float cp = cbuf[idx];
            float hp = hcur[idx];                     // previous h for zeta term
            float cn = sigm(fg) * cp + sigm(ig) * tanhf(gg);
            float hn = sigm(og) * tanhf(cn);
            float ho = hn + ZW * wt * hp;
            cbuf[idx] = cn;
            hnxt[idx] = ho;
            out[((size_t)b * Tc + t) * Hc + colL] = ho;
            if (t == Tc - 1) {                        // h_f, c_f tails
                out[(size_t)Bc * Tc * Hc + idx] = ho;
                out[(size_t)Bc * Tc * Hc + (size_t)Bc * Hc + idx] = cn;
            }
        }

        // Prefetch next step's scattered x_proj operands (global_prefetch_b8,
        // counter-free) so they resolve while we spin in the grid barrier.
        if (xp != nullptr && t + 1 < Tc) {
#pragma unroll
            for (int g = 0; g < 4; ++g)
#pragma unroll
                for (int v = 0; v < 8; ++v) {
                    int b = mt * 16 + v + 8 * lh;
                    __builtin_prefetch(xp + ((size_t)b * Tc + (t + 1)) * G4 + g * Hc + colL,
                                       0, 1);
                }
        }

        tgt += NBLK;
        grid_barrier(bar, tgt);   // publish h(t+1), retire h(t)
    }
}

// ---------------- host launcher --------------------------------------------

extern "C" void kernel_launch(void* const* d_in, const int* in_sizes, int n_in,
                              void* d_out, int out_size, void* d_ws, size_t ws_size,
                              hipStream_t stream) {
    (void)in_sizes; (void)n_in; (void)out_size;
    const float* x       = (const float*)d_in[0];
    const float* Wih     = (const float*)d_in[1];
    const float* Whh     = (const float*)d_in[2];
    const float* bih     = (const float*)d_in[3];
    const float* bhh     = (const float*)d_in[4];
    const float* log_phi = (const float*)d_in[5];
    const float* gammas  = (const float*)d_in[6];
    float* out = (float*)d_out;
    float* ws  = (float*)d_ws;

    // workspace layout (float offsets)
    unsigned* bar = (unsigned*)ws;                  // [0,64)    barrier counter
    float* wts  = ws + 64;                          // 1024      zeta weights
    float* hA   = ws + 2048;                        // 32768     h ping
    float* hB   = hA + Bc * Hc;                     // 32768     h pong
    float* cbuf = hB + Bc * Hc;                     // 32768     cell state
    float* xp   = ws + 131072;                      // 64*1024*2048 x-projection

    size_t need_xp = (size_t)131072 * sizeof(float) +
                     (size_t)Bc * Tc * G4 * sizeof(float);  // ~537 MB
    const bool use_xp = (ws_size >= need_xp);       // constant per process -> deterministic

    hipMemsetAsync(bar, 0, 256, stream);
    hipMemsetAsync(hA, 0, (size_t)Bc * Hc * sizeof(float), stream);
    hipMemsetAsync(cbuf, 0, (size_t)Bc * Hc * sizeof(float), stream);

    k_wts<<<dim3((Tc + 255) / 256), dim3(256), 0, stream>>>(log_phi, gammas, wts);

    if (use_xp) {
        k_xproj<<<dim3(Bc * Tc / 64, G4 / 128), dim3(256), 0, stream>>>(x, Wih, bih, bhh, xp);
    }
    k_scan<<<dim3(NBLK), dim3(512), 0, stream>>>(use_xp ? xp : nullptr, x, Wih, Whh,
                                                 bih, bhh, wts, hA, hB, cbuf, out, bar);
}